// Net_58978490909308
// MI455X (gfx1250) — compile-verified
//
#include <hip/hip_runtime.h>
#include <hip/hip_bf16.h>

typedef float v2f __attribute__((ext_vector_type(2)));
typedef float v8f __attribute__((ext_vector_type(8)));

#define NUM_NODES 100000
#define NUM_EDGES 3200000
#define NFEAT     512
#define D1        16
#define D2        8
#define NTILES    (NUM_NODES / 16)   // 6250 exactly

// ---------------- degree / normalization ----------------

__global__ void k_deg_init(float* __restrict__ deg) {
    int n = blockIdx.x * blockDim.x + threadIdx.x;
    if (n < NUM_NODES) deg[n] = 1.0f;          // self loop
}

__global__ void k_deg_edges(const long long* __restrict__ dst, float* __restrict__ deg) {
    int e = blockIdx.x * blockDim.x + threadIdx.x;
    if (e < NUM_EDGES) atomicAdd(&deg[dst[e]], 1.0f);
}

__global__ void k_deg_to_dis(float* __restrict__ deg) {
    int n = blockIdx.x * blockDim.x + threadIdx.x;
    if (n < NUM_NODES) deg[n] = rsqrtf(deg[n]);   // deg >= 1 always
}

// ---------------- GEMM1 via WMMA f32 16x16x4 ----------------
// hs1[n][j] = (x @ W1)[n][j] * dis[n]   ; agg1 initialized with the self-loop term hs1.

__global__ __launch_bounds__(256) void k_gemm1_wmma(
    const float* __restrict__ x, const float* __restrict__ W1,
    const float* __restrict__ dis,
    float* __restrict__ hs1, float* __restrict__ agg1)
{
    __shared__ float w[NFEAT * D1];            // 32 KB
    {   // cooperative load of W1 into LDS as float4
        const float4* wsrc = (const float4*)W1;
        float4* wdst = (float4*)w;
        for (int i = threadIdx.x; i < (NFEAT * D1) / 4; i += 256) wdst[i] = wsrc[i];
    }
    __syncthreads();

    const int wave = threadIdx.x >> 5;
    const int lane = threadIdx.x & 31;
    const int tile = blockIdx.x * 8 + wave;
    if (tile >= NTILES) return;                // whole-wave uniform exit; EXEC stays all-1s

    const int col = lane & 15;                 // A-row / B-col / D-col
    const int hi  = lane >> 4;                 // 0 or 1
    const int kb  = hi * 2;                    // K sub-offset for A/B fragments
    const float* __restrict__ xrow = x + (size_t)(tile * 16 + col) * NFEAT;

    v8f acc = {};
    #pragma unroll 4
    for (int k = 0; k < NFEAT; k += 4) {
        v2f a, b;
        a.x = xrow[k + kb];
        a.y = xrow[k + kb + 1];
        b.x = w[(k + kb)     * D1 + col];
        b.y = w[(k + kb + 1) * D1 + col];
        // D = A(16x4 f32) * B(4x16 f32) + C
        acc = __builtin_amdgcn_wmma_f32_16x16x4_f32(
                /*neg_a=*/false, a, /*neg_b=*/false, b,
                /*c_mod=*/(short)0, acc, /*reuse_a=*/false, /*reuse_b=*/false);
    }

    #pragma unroll
    for (int v = 0; v < 8; ++v) {
        int m = tile * 16 + v + 8 * hi;        // D row for VGPR v
        float val = acc[v] * dis[m];           // pre-scale by dis[src-side]
        hs1 [(size_t)m * D1 + col] = val;
        agg1[(size_t)m * D1 + col] = val;      // self-loop contribution
    }
}

// ---------------- edge scatter, layer 1 (16 features, 4 lanes/edge) ----------------

__global__ void k_scatter1(const long long* __restrict__ src,
                           const long long* __restrict__ dst,
                           const float* __restrict__ hs1, float* __restrict__ agg1)
{
    unsigned gid = blockIdx.x * blockDim.x + threadIdx.x;
    unsigned e = gid >> 2;
    if (e >= NUM_EDGES) return;
    unsigned q = (gid & 3u) * 4u;
    long long s = src[e], d = dst[e];
    float4 v = *(const float4*)(hs1 + (size_t)s * D1 + q);
    float* o = agg1 + (size_t)d * D1 + q;
    atomicAdd(o + 0, v.x);
    atomicAdd(o + 1, v.y);
    atomicAdd(o + 2, v.z);
    atomicAdd(o + 3, v.w);
}

// ---------------- fused: relu1 + GEMM2 + dis prescale + agg2 init ----------------

__global__ __launch_bounds__(256) void k_mid(
    const float* __restrict__ agg1, const float* __restrict__ dis,
    const float* __restrict__ b1,   const float* __restrict__ W2,
    float* __restrict__ hs2, float* __restrict__ agg2)
{
    __shared__ float w2[D1 * D2];   // 128 floats
    __shared__ float sb1[D1];
    if (threadIdx.x < D1 * D2) w2[threadIdx.x] = W2[threadIdx.x];
    if (threadIdx.x < D1)      sb1[threadIdx.x] = b1[threadIdx.x];
    __syncthreads();

    int n = blockIdx.x * 256 + threadIdx.x;
    if (n >= NUM_NODES) return;
    float dn = dis[n];

    float h[D1];
    #pragma unroll
    for (int j = 0; j < D1; ++j)
        h[j] = fmaxf(dn * agg1[(size_t)n * D1 + j] + sb1[j], 0.0f);

    float o[D2];
    #pragma unroll
    for (int c = 0; c < D2; ++c) o[c] = 0.0f;
    #pragma unroll
    for (int j = 0; j < D1; ++j) {
        #pragma unroll
        for (int c = 0; c < D2; ++c) o[c] += h[j] * w2[j * D2 + c];
    }
    #pragma unroll
    for (int c = 0; c < D2; ++c) {
        float v = o[c] * dn;                   // pre-scale by dis[n]
        hs2 [(size_t)n * D2 + c] = v;
        agg2[(size_t)n * D2 + c] = v;          // self-loop contribution
    }
}

// ---------------- edge scatter, layer 2 (8 features, 2 lanes/edge) ----------------

__global__ void k_scatter2(const long long* __restrict__ src,
                           const long long* __restrict__ dst,
                           const float* __restrict__ hs2, float* __restrict__ agg2)
{
    unsigned gid = blockIdx.x * blockDim.x + threadIdx.x;
    unsigned e = gid >> 1;
    if (e >= NUM_EDGES) return;
    unsigned q = (gid & 1u) * 4u;
    long long s = src[e], d = dst[e];
    float4 v = *(const float4*)(hs2 + (size_t)s * D2 + q);
    float* o = agg2 + (size_t)d * D2 + q;
    atomicAdd(o + 0, v.x);
    atomicAdd(o + 1, v.y);
    atomicAdd(o + 2, v.z);
    atomicAdd(o + 3, v.w);
}

// ---------------- final: relu2 + linear head ----------------

__global__ void k_final(const float* __restrict__ agg2, const float* __restrict__ dis,
                        const float* __restrict__ b2, const float* __restrict__ Wl,
                        const float* __restrict__ bl, float* __restrict__ out)
{
    int n = blockIdx.x * blockDim.x + threadIdx.x;
    if (n >= NUM_NODES) return;
    float dn = dis[n];
    float acc = bl[0];
    #pragma unroll
    for (int c = 0; c < D2; ++c)
        acc += fmaxf(dn * agg2[(size_t)n * D2 + c] + b2[c], 0.0f) * Wl[c];
    out[n] = acc;
}

// ---------------- launcher ----------------

extern "C" void kernel_launch(void* const* d_in, const int* in_sizes, int n_in,
                              void* d_out, int out_size, void* d_ws, size_t ws_size,
                              hipStream_t stream) {
    const float*     x    = (const float*)d_in[0];
    const long long* ei   = (const long long*)d_in[1];   // int64, shape (2, E) flat
    const float*     W1   = (const float*)d_in[2];
    const float*     b1   = (const float*)d_in[3];
    const float*     W2   = (const float*)d_in[4];
    const float*     b2   = (const float*)d_in[5];
    const float*     Wl   = (const float*)d_in[6];
    const float*     bl   = (const float*)d_in[7];
    float*           out  = (float*)d_out;

    const long long* src = ei;
    const long long* dst = ei + NUM_EDGES;

    // workspace layout (floats), 256-aligned blocks; total ~19.6 MB
    float* ws   = (float*)d_ws;
    float* dis  = ws;                            // 100000 -> pad to 100352
    float* hs1  = ws + 100352;                   // N*16
    float* agg1 = hs1 + (size_t)NUM_NODES * D1;  // N*16
    float* hs2  = agg1 + (size_t)NUM_NODES * D1; // N*8
    float* agg2 = hs2 + (size_t)NUM_NODES * D2;  // N*8

    const int nodeBlocks = (NUM_NODES + 255) / 256;          // 391
    const int edgeBlocks = (NUM_EDGES + 255) / 256;          // 12500

    // 1) degree + dis = rsqrt(deg)
    k_deg_init<<<nodeBlocks, 256, 0, stream>>>(dis);
    k_deg_edges<<<edgeBlocks, 256, 0, stream>>>(dst, dis);
    k_deg_to_dis<<<nodeBlocks, 256, 0, stream>>>(dis);

    // 2) hs1 = (x @ W1) * dis ; agg1 = hs1 (self-loop)
    k_gemm1_wmma<<<(NTILES + 7) / 8, 256, 0, stream>>>(x, W1, dis, hs1, agg1);

    // 3) agg1[dst] += hs1[src]
    k_scatter1<<<(NUM_EDGES * 4 + 255) / 256, 256, 0, stream>>>(src, dst, hs1, agg1);

    // 4) h1 = relu(dis*agg1 + b1); hs2 = (h1 @ W2) * dis; agg2 = hs2
    k_mid<<<nodeBlocks, 256, 0, stream>>>(agg1, dis, b1, W2, hs2, agg2);

    // 5) agg2[dst] += hs2[src]
    k_scatter2<<<(NUM_EDGES * 2 + 255) / 256, 256, 0, stream>>>(src, dst, hs2, agg2);

    // 6) out = relu(dis*agg2 + b2) @ Wl + bl
    k_final<<<nodeBlocks, 256, 0, stream>>>(agg2, dis, b2, Wl, bl, out);
}